// GraphGRU_661424963763
// MI455X (gfx1250) — compile-verified
//
#include <hip/hip_runtime.h>
#include <hip/hip_bf16.h>

// ---------------------------------------------------------------------------
// GraphGRU on MI455X (gfx1250, wave32).
//
// Per-role pre-aggregation of raw features (L2-resident scatter), then stacked
// fp32 WMMA GEMMs (V_WMMA_F32_16X16X4_F32) fed by GLOBAL_LOAD_ASYNC_TO_LDS
// (activations, ASYNCcnt) and the Tensor Data Mover (weights, TENSORcnt),
// then elementwise GRU fusion. Weights are stored K-pair interleaved so B
// fragments are single contiguous ds_load_b64s.
// ---------------------------------------------------------------------------

typedef float v2f __attribute__((ext_vector_type(2)));
typedef float v8f __attribute__((ext_vector_type(8)));
typedef unsigned int u32x4 __attribute__((ext_vector_type(4)));
typedef unsigned int u32x8 __attribute__((ext_vector_type(8)));

#define KDIM 768   // 3 role-aggregates (384) + 3 role-masked self segments (384)

// Generic -> LDS(addrspace 3) offset for async/TDM destinations.
__device__ __forceinline__ unsigned int lds_off(void* p) {
  return (unsigned int)(unsigned long long)(__attribute__((address_space(3))) void*)p;
}

// Per-lane async 16B global->LDS copy (CDNA5 VGLOBAL async op, ASYNCcnt).
__device__ __forceinline__ void async_copy_b128(const void* gsrc, void* ldst) {
  asm volatile("global_load_async_to_lds_b128 %0, %1, off"
               :: "v"(lds_off(ldst)), "v"((unsigned long long)gsrc)
               : "memory");
}

__device__ __forceinline__ void wait_asynccnt0() {
  asm volatile("s_wait_asynccnt 0x0" ::: "memory");
}

// ---------------------------------------------------------------------------
// Tensor Data Mover: DMA a 2D fp32 tile (tile_w x tile_h, row stride
// stride_elems) from global memory into LDS at lds_addr, with LDS row padding
// pad_amount(code 7 -> 8 DWORDs) every pad_interval(code 6 -> 128 DWORDs), so
// a 128-float row lands at stride 136 floats. D# per CDNA5 ISA 8.3/8.4.
// Tensor dims == tile dims (descriptor base == tile start, fully in-bounds).
// ---------------------------------------------------------------------------
__device__ __forceinline__ void tdm_load_tile_f32(
    const void* gaddr, unsigned int lds_addr,
    unsigned int tile_w, unsigned int tile_h, unsigned int stride_elems) {
  unsigned long long ga = (unsigned long long)gaddr;
  u32x4 g0;
  g0[0] = 1u;                                             // count=1 (valid user D#)
  g0[1] = lds_addr;                                       // LDS byte address
  g0[2] = (unsigned int)ga;                               // global_addr[31:0]
  g0[3] = ((unsigned int)(ga >> 32) & 0x01FFFFFFu)        // global_addr[56:32]
          | (2u << 30);                                   // type=2 ("image")
  u32x8 g1;
  g1[0] = (2u << 16)        // data_size = 4 bytes
        | (1u << 20)        // pad_enable
        | (6u << 22)        // pad_interval code 6 = 128 DWORDs
        | (7u << 25);       // pad_amount  code 7 = 8 DWORDs
  g1[1] = (tile_w & 0xFFFFu) << 16;        // tensor_dim0[15:0]
  g1[2] = (tile_h & 0xFFFFu) << 16;        // tensor_dim0[31:16]=0 | tensor_dim1[15:0]
  g1[3] = (tile_w & 0xFFFFu) << 16;        // tensor_dim1[31:16]=0 | tile_dim0
  g1[4] = tile_h & 0xFFFFu;                // tile_dim1 | tile_dim2=0
  g1[5] = stride_elems;                    // tensor_dim0_stride[31:0] (elements)
  g1[6] = 0;                               // stride[47:32] | dim1_stride[15:0]
  g1[7] = 0;                               // dim1_stride[47:16]
  asm volatile("tensor_load_to_lds %0, %1" :: "s"(g0), "s"(g1) : "memory");
}

__device__ __forceinline__ void wait_tensorcnt0() {
  __builtin_amdgcn_s_wait_tensorcnt(0);
}

// ---------------------------------------------------------------------------
// Edge aggregation: for each edge (row->col), add feat[row] into
// U[col][ role(row)*128 .. +128 ]. One wave (32 lanes) per edge, 4 floats/lane.
// Self-loops (appended range) included. Optionally counts per-role in-degree.
// Working set (U 154MB + feat 26MB) fits the 192MB L2 -> atomics stay on-chip.
// ---------------------------------------------------------------------------
__global__ __launch_bounds__(256) void edge_aggregate_k(
    const float* __restrict__ feat, const int* __restrict__ ei,
    const int* __restrict__ role, float* __restrict__ U,
    float* __restrict__ cnt, int E, int N) {
  long long w = ((long long)blockIdx.x * blockDim.x + threadIdx.x) >> 5;
  int lane = threadIdx.x & 31;
  if (w >= (long long)E + N) return;
  int row, col;
  if (w < E) { row = ei[w]; col = ei[(long long)E + w]; }
  else       { row = col = (int)(w - E); }
  int r = role[row];
  const float4 v = ((const float4*)feat)[(size_t)row * 32 + lane];
  float* dst = U + (size_t)col * KDIM + r * 128 + lane * 4;
  atomicAdd(dst + 0, v.x);
  atomicAdd(dst + 1, v.y);
  atomicAdd(dst + 2, v.z);
  atomicAdd(dst + 3, v.w);
  if (cnt != nullptr && lane == 0) atomicAdd(cnt + (size_t)col * 3 + r, 1.0f);
}

// ---------------------------------------------------------------------------
// Normalize aggregates by in-degree and fill role-masked self segments for
// both Ux (from x) and Uh (from h_prev). Also stores w_r = cnt_r/cnt and 1/cnt.
// ---------------------------------------------------------------------------
__global__ __launch_bounds__(128) void normalize_xh_k(
    const float* __restrict__ x, const float* __restrict__ h,
    const int* __restrict__ role, const float* __restrict__ cnt,
    float* __restrict__ Ux, float* __restrict__ Uh,
    float* __restrict__ w4, int N) {
  int i = blockIdx.x;
  int d = threadIdx.x;
  if (i >= N) return;
  float c0 = cnt[(size_t)i*3+0], c1 = cnt[(size_t)i*3+1], c2 = cnt[(size_t)i*3+2];
  float inv = 1.0f / fmaxf(c0 + c1 + c2, 1.0f);
  int ro = role[i];
  size_t base = (size_t)i * KDIM;
#pragma unroll
  for (int r = 0; r < 3; ++r) {
    Ux[base + r*128 + d] *= inv;
    Uh[base + r*128 + d] *= inv;
  }
  Ux[base + 384 + ro*128 + d] = x[(size_t)i*128 + d];
  Uh[base + 384 + ro*128 + d] = h[(size_t)i*128 + d];
  if (d == 0) {
    w4[(size_t)i*4+0] = c0*inv; w4[(size_t)i*4+1] = c1*inv;
    w4[(size_t)i*4+2] = c2*inv; w4[(size_t)i*4+3] = inv;
  }
}

__global__ __launch_bounds__(128) void normalize_s_k(
    const float* __restrict__ s, const int* __restrict__ role,
    const float* __restrict__ w4, float* __restrict__ Us, int N) {
  int i = blockIdx.x;
  int d = threadIdx.x;
  if (i >= N) return;
  float inv = w4[(size_t)i*4+3];
  size_t base = (size_t)i * KDIM;
#pragma unroll
  for (int r = 0; r < 3; ++r) Us[base + r*128 + d] *= inv;
  Us[base + 384 + role[i]*128 + d] = s[(size_t)i*128 + d];
}

// ---------------------------------------------------------------------------
// Tiny precompute kernels: role gates and stacked weight matrices.
//   logical row k in [0,384):   W * diag(sigmoid(Wg[r]+bg[r]))  (aggregate)
//   logical row k in [384,768): T_r = W @ S[r]                  (masked self)
// Stored K-pair interleaved: element (k, c) -> Mdst[(k>>1)*2*ldM + 2c + (k&1)]
// so WMMA B fragments {B[k][n], B[k+1][n]} are contiguous 8 bytes.
// ---------------------------------------------------------------------------
__global__ void build_gate_k(const float* __restrict__ Wg,
                             const float* __restrict__ bg,
                             float* __restrict__ g) {
  int t = threadIdx.x;  // 384 = 3*128
  g[t] = 1.0f / (1.0f + expf(-(Wg[t] + bg[t])));
}

__global__ __launch_bounds__(256) void build_M_k(
    const float* __restrict__ W, const float* __restrict__ S,
    const float* __restrict__ g, float* __restrict__ Mdst,
    int ldM, int colOff) {
  int idx = blockIdx.x * blockDim.x + threadIdx.x;  // 768*128
  if (idx >= KDIM * 128) return;
  int row = idx >> 7, d = idx & 127;
  float val;
  if (row < 384) {
    int r = row >> 7, k = row & 127;
    val = W[k*128 + d] * g[r*128 + d];
  } else {
    int rr = row - 384;
    int r = rr >> 7, k = rr & 127;
    const float* Sp = S + (size_t)r * 128 * 128;
    float acc = 0.0f;
    for (int j = 0; j < 128; ++j) acc = fmaf(W[k*128 + j], Sp[j*128 + d], acc);
    val = acc;
  }
  int c = colOff + d;
  Mdst[(size_t)(row >> 1) * (2 * ldM) + 2 * c + (row & 1)] = val;
}

// ---------------------------------------------------------------------------
// fp32 WMMA GEMM: C[Nrows x Ncols] = U[Nrows x K] @ Mw[K x Ncols]
// Mw is K-pair interleaved: (K/2) x (2*Ncols).
// Block: 256 threads (8 waves), tile 128x64, K staged in 32-chunks via LDS.
// U tile: GLOBAL_LOAD_ASYNC_TO_LDS_B128 (per-lane, ASYNCcnt).
// Weight tile: TENSOR_LOAD_TO_LDS (one TDM descriptor per block per chunk,
// TENSORcnt): 128 wide x 16 K-pair rows, TDM row padding -> LDS stride 136.
// Wave grid 4x2; each wave owns a 32x32 patch = 2x2 WMMA 16x16 f32 tiles.
// Fragment layouts per CDNA5 ISA 7.12.2 (wave32):
//   A 16x4 : v0 = A[l16][k + 2*half],  v1 = A[l16][k + 2*half + 1]
//   B 4x16 : v0 = B[k + 2*half][l16],  v1 = B[k + 2*half + 1][l16]
//   C 16x16: vgpr v = C[v + 8*half][l16]
// ---------------------------------------------------------------------------
__global__ __launch_bounds__(256, 2) void wmma_gemm_f32_k(
    const float* __restrict__ U, const float* __restrict__ Mw,
    float* __restrict__ C, int Nrows, int K, int Ncols) {
  __shared__ float Ut[128][36];    // [row][k], pad 36 floats (144B, 16B aligned)
  __shared__ float Mrow[16][136];  // [kpair][2n+(k&1)], TDM-padded rows

  const int tid  = threadIdx.x;
  const int wave = tid >> 5;
  const int lane = tid & 31;
  const int half = lane >> 4;
  const int l16  = lane & 15;
  const int wm   = wave & 3;   // 4 row quarters
  const int wn   = wave >> 2;  // 2 col halves
  const int rowBase = blockIdx.y * 128;
  const int colBase = blockIdx.x * 64;

  v8f acc[2][2] = {};

  for (int k0 = 0; k0 < K; k0 += 32) {
    // one TDM descriptor per block: interleaved weight tile 128 x 16 pairs
    if (wave == 0) {
      tdm_load_tile_f32(Mw + (size_t)(k0 >> 1) * (2 * Ncols) + 2 * colBase,
                        lds_off(&Mrow[0][0]), 128u, 16u, (unsigned)(2 * Ncols));
    }
    // stage U tile 128x32 via async global->LDS (one b128 per thread x4)
#pragma unroll
    for (int i = 0; i < 4; ++i) {
      int idx = tid + i * 256;       // 0..1023
      int r   = idx >> 3;
      int c4  = (idx & 7) << 2;
      int gr  = rowBase + r;
      if (gr >= Nrows) gr = Nrows - 1;  // clamp; tail rows never stored
      async_copy_b128(U + (size_t)gr * K + k0 + c4, &Ut[r][c4]);
    }
    wait_asynccnt0();                  // our async copies landed in LDS
    if (wave == 0) wait_tensorcnt0();  // TDM tile landed in LDS
    __syncthreads();                   // publish to all waves

#pragma unroll
    for (int kk = 0; kk < 32; kk += 4) {
      const int kidx = kk + half * 2;  // always even
      v2f a[2], b[2];
#pragma unroll
      for (int tm = 0; tm < 2; ++tm) {
        int m = wm * 32 + tm * 16 + l16;
        a[tm] = (v2f){Ut[m][kidx], Ut[m][kidx + 1]};
      }
#pragma unroll
      for (int tn = 0; tn < 2; ++tn) {
        int n = wn * 32 + tn * 16 + l16;
        b[tn] = *(const v2f*)&Mrow[kidx >> 1][2 * n];  // contiguous b64
      }
#pragma unroll
      for (int tm = 0; tm < 2; ++tm)
#pragma unroll
        for (int tn = 0; tn < 2; ++tn)
          acc[tm][tn] = __builtin_amdgcn_wmma_f32_16x16x4_f32(
              false, a[tm], false, b[tn], (short)0, acc[tm][tn], false, false);
    }
    __syncthreads();
  }

  // store C (coalesced: lanes 0..15 hit consecutive columns)
#pragma unroll
  for (int tm = 0; tm < 2; ++tm) {
#pragma unroll
    for (int tn = 0; tn < 2; ++tn) {
      int col = colBase + wn * 32 + tn * 16 + l16;
#pragma unroll
      for (int v = 0; v < 8; ++v) {
        int row = rowBase + wm * 32 + tm * 16 + half * 8 + v;
        if (row < Nrows) C[(size_t)row * Ncols + col] = acc[tm][tn][v];
      }
    }
  }
}

// ---------------------------------------------------------------------------
// GRU fusion kernels.
// ---------------------------------------------------------------------------
__global__ __launch_bounds__(256) void fusion_zr_k(
    const float* __restrict__ Zx, const float* __restrict__ Zh,
    const float* __restrict__ h_prev, const float* __restrict__ w4,
    const float* __restrict__ b_xz, const float* __restrict__ b_xr,
    float* __restrict__ zbuf, float* __restrict__ sbuf, int N) {
  int idx = blockIdx.x * blockDim.x + threadIdx.x;
  if (idx >= N * 128) return;
  int i = idx >> 7, d = idx & 127;
  float w0 = w4[(size_t)i*4], w1 = w4[(size_t)i*4+1], w2 = w4[(size_t)i*4+2];
  float bz = w0*b_xz[d] + w1*b_xz[128+d] + w2*b_xz[256+d];
  float br = w0*b_xr[d] + w1*b_xr[128+d] + w2*b_xr[256+d];
  float axz = fmaxf(Zx[(size_t)i*384 + d]        + bz, 0.0f);
  float ahz = fmaxf(Zh[(size_t)i*256 + d],             0.0f);
  float axr = fmaxf(Zx[(size_t)i*384 + 128 + d]  + br, 0.0f);
  float ahr = fmaxf(Zh[(size_t)i*256 + 128 + d],       0.0f);
  float z = 1.0f / (1.0f + expf(-(axz + ahz)));
  float r = 1.0f / (1.0f + expf(-(axr + ahr)));
  zbuf[idx] = z;
  sbuf[idx] = r * h_prev[idx];
}

__global__ __launch_bounds__(256) void fusion_h_k(
    const float* __restrict__ Zx, const float* __restrict__ Zhh,
    const float* __restrict__ h_prev, const float* __restrict__ w4,
    const float* __restrict__ b_xh, const float* __restrict__ zbuf,
    float* __restrict__ out, int N) {
  int idx = blockIdx.x * blockDim.x + threadIdx.x;
  if (idx >= N * 128) return;
  int i = idx >> 7, d = idx & 127;
  float w0 = w4[(size_t)i*4], w1 = w4[(size_t)i*4+1], w2 = w4[(size_t)i*4+2];
  float bh = w0*b_xh[d] + w1*b_xh[128+d] + w2*b_xh[256+d];
  float axh = fmaxf(Zx[(size_t)i*384 + 256 + d] + bh, 0.0f);
  float ahh = fmaxf(Zhh[(size_t)i*128 + d],           0.0f);
  float ht = tanhf(axh + ahh);
  float z  = zbuf[idx];
  out[idx] = z * h_prev[idx] + (1.0f - z) * ht;
}

// ---------------------------------------------------------------------------
// Host orchestration.
// ---------------------------------------------------------------------------
extern "C" void kernel_launch(void* const* d_in, const int* in_sizes, int n_in,
                              void* d_out, int out_size, void* d_ws, size_t ws_size,
                              hipStream_t stream) {
  const float* x      = (const float*)d_in[0];
  const float* h_prev = (const float*)d_in[1];
  const int*   ei     = (const int*)d_in[2];
  const int*   role   = (const int*)d_in[3];
  const int N = in_sizes[0] / 128;
  const int E = in_sizes[2] / 2;

  // per-conv inputs in setup_inputs dict order
  const float* xz_W  = (const float*)d_in[4];
  const float* xz_Wg = (const float*)d_in[5];
  const float* xz_bg = (const float*)d_in[6];
  const float* xz_S  = (const float*)d_in[7];
  const float* xz_b  = (const float*)d_in[8];
  const float* hz_W  = (const float*)d_in[9];
  const float* hz_Wg = (const float*)d_in[10];
  const float* hz_bg = (const float*)d_in[11];
  const float* hz_S  = (const float*)d_in[12];
  const float* xr_W  = (const float*)d_in[13];
  const float* xr_Wg = (const float*)d_in[14];
  const float* xr_bg = (const float*)d_in[15];
  const float* xr_S  = (const float*)d_in[16];
  const float* xr_b  = (const float*)d_in[17];
  const float* hr_W  = (const float*)d_in[18];
  const float* hr_Wg = (const float*)d_in[19];
  const float* hr_bg = (const float*)d_in[20];
  const float* hr_S  = (const float*)d_in[21];
  const float* xh_W  = (const float*)d_in[22];
  const float* xh_Wg = (const float*)d_in[23];
  const float* xh_bg = (const float*)d_in[24];
  const float* xh_S  = (const float*)d_in[25];
  const float* xh_b  = (const float*)d_in[26];
  const float* hh_W  = (const float*)d_in[27];
  const float* hh_Wg = (const float*)d_in[28];
  const float* hh_bg = (const float*)d_in[29];
  const float* hh_S  = (const float*)d_in[30];

  // workspace carve-up
  char* ws = (char*)d_ws;
  auto carve = [&](size_t bytes) -> char* {
    char* p = ws;
    ws += (bytes + 255) & ~(size_t)255;
    return p;
  };
  float* Ux   = (float*)carve((size_t)N * KDIM * 4);  // 153.6 MB
  float* Uh   = (float*)carve((size_t)N * KDIM * 4);  // reused as Us later
  float* Zx   = (float*)carve((size_t)N * 384 * 4);
  float* Zh   = (float*)carve((size_t)N * 256 * 4);   // reused as Zhh later
  float* zbuf = (float*)carve((size_t)N * 128 * 4);
  float* sbuf = (float*)carve((size_t)N * 128 * 4);
  float* cnt  = (float*)carve((size_t)N * 3 * 4);
  float* w4   = (float*)carve((size_t)N * 4 * 4);
  float* gbuf = (float*)carve((size_t)6 * 384 * 4);
  float* Mx   = (float*)carve((size_t)KDIM * 384 * 4);
  float* Mh   = (float*)carve((size_t)KDIM * 256 * 4);
  float* Mhh  = (float*)carve((size_t)KDIM * 128 * 4);

  // 1) zero the atomic accumulators
  hipMemsetAsync(Ux, 0, (size_t)N * KDIM * 4, stream);
  hipMemsetAsync(Uh, 0, (size_t)N * KDIM * 4, stream);
  hipMemsetAsync(cnt, 0, (size_t)N * 3 * 4, stream);

  // 2) role gates: sigmoid(Wg + bg), 3x128 per conv (order: xz hz xr hr xh hh)
  build_gate_k<<<1, 384, 0, stream>>>(xz_Wg, xz_bg, gbuf + 0*384);
  build_gate_k<<<1, 384, 0, stream>>>(hz_Wg, hz_bg, gbuf + 1*384);
  build_gate_k<<<1, 384, 0, stream>>>(xr_Wg, xr_bg, gbuf + 2*384);
  build_gate_k<<<1, 384, 0, stream>>>(hr_Wg, hr_bg, gbuf + 3*384);
  build_gate_k<<<1, 384, 0, stream>>>(xh_Wg, xh_bg, gbuf + 4*384);
  build_gate_k<<<1, 384, 0, stream>>>(hh_Wg, hh_bg, gbuf + 5*384);

  // 3) stacked weight matrices, K-pair interleaved (K/2 x 2*Ncols)
  const int mblk = (KDIM * 128 + 255) / 256;
  build_M_k<<<mblk, 256, 0, stream>>>(xz_W, xz_S, gbuf + 0*384, Mx, 384, 0);
  build_M_k<<<mblk, 256, 0, stream>>>(xr_W, xr_S, gbuf + 2*384, Mx, 384, 128);
  build_M_k<<<mblk, 256, 0, stream>>>(xh_W, xh_S, gbuf + 4*384, Mx, 384, 256);
  build_M_k<<<mblk, 256, 0, stream>>>(hz_W, hz_S, gbuf + 1*384, Mh, 256, 0);
  build_M_k<<<mblk, 256, 0, stream>>>(hr_W, hr_S, gbuf + 3*384, Mh, 256, 128);
  build_M_k<<<mblk, 256, 0, stream>>>(hh_W, hh_S, gbuf + 5*384, Mhh, 128, 0);

  // 4-5) edge aggregation (one wave per edge; self-loops appended)
  {
    long long waves = (long long)E + N;
    int blocks = (int)((waves * 32 + 255) / 256);
    edge_aggregate_k<<<blocks, 256, 0, stream>>>(x,      ei, role, Ux, cnt,    E, N);
    edge_aggregate_k<<<blocks, 256, 0, stream>>>(h_prev, ei, role, Uh, nullptr, E, N);
  }

  // 6) normalize + fill role-masked self segments
  normalize_xh_k<<<N, 128, 0, stream>>>(x, h_prev, role, cnt, Ux, Uh, w4, N);

  // 7-8) stacked GEMMs: Zx = Ux@Mx (xz|xr|xh), Zh = Uh@Mh (hz|hr)
  {
    dim3 blk(256);
    dim3 gx(384 / 64, (N + 127) / 128);
    wmma_gemm_f32_k<<<gx, blk, 0, stream>>>(Ux, Mx, Zx, N, KDIM, 384);
    dim3 gh(256 / 64, (N + 127) / 128);
    wmma_gemm_f32_k<<<gh, blk, 0, stream>>>(Uh, Mh, Zh, N, KDIM, 256);
  }

  // 9) z, r gates; s = r * h_prev
  {
    int blocks = (N * 128 + 255) / 256;
    fusion_zr_k<<<blocks, 256, 0, stream>>>(Zx, Zh, h_prev, w4, xz_b, xr_b,
                                            zbuf, sbuf, N);
  }

  // 10-12) aggregate s (reuse Uh as Us), normalize, GEMM Zhh = Us@Mhh
  hipMemsetAsync(Uh, 0, (size_t)N * KDIM * 4, stream);
  {
    long long waves = (long long)E + N;
    int blocks = (int)((waves * 32 + 255) / 256);
    edge_aggregate_k<<<blocks, 256, 0, stream>>>(sbuf, ei, role, Uh, nullptr, E, N);
  }
  normalize_s_k<<<N, 128, 0, stream>>>(sbuf, role, w4, Uh, N);
  {
    dim3 blk(256);
    dim3 g(128 / 64, (N + 127) / 128);
    wmma_gemm_f32_k<<<g, blk, 0, stream>>>(Uh, Mhh, Zh, N, KDIM, 128);  // Zh reused as Zhh
  }

  // 13) h_tilde and final GRU update
  {
    int blocks = (N * 128 + 255) / 256;
    fusion_h_k<<<blocks, 256, 0, stream>>>(Zx, Zh, h_prev, w4, xh_b, zbuf,
                                           (float*)d_out, N);
  }
}